// AggregatedInfluenceScorer_84877143703614
// MI455X (gfx1250) — compile-verified
//
#include <hip/hip_runtime.h>
#include <hip/hip_bf16.h>

typedef __attribute__((ext_vector_type(2))) float v2f;
typedef __attribute__((ext_vector_type(8))) float v8f;

__device__ __forceinline__ float fast_tanhf(float x) {
#if __has_builtin(__builtin_amdgcn_tanhf)
  return __builtin_amdgcn_tanhf(x);       // gfx1250 v_tanh_f32 (TRANS)
#else
  float y;
  asm("v_tanh_f32 %0, %1" : "=v"(y) : "v"(x));
  return y;
#endif
}

// C[M,N] = A[M,K] @ B[K,N] + bias[N], all fp32 row-major.
// One 16x16 output tile per wave using V_WMMA_F32_16X16X4_F32.
// N,K compile-time: all loads become base + immediate offset (no per-iter
// pointer arithmetic, no remainder loop).
// A 16x4 layout : lane l, reg j -> A[m = l%16][k = 2*(l/16) + j]
// B 4x16 layout : lane l, reg j -> B[k = 2*(l/16) + j][n = l%16]
// C/D 16x16     : lane l, reg r -> C[m = r + 8*(l/16)][n = l%16]
template <int N, int K>
__global__ __launch_bounds__(128) void proj_gemm_wmma(
    const float* __restrict__ A, const float* __restrict__ B,
    const float* __restrict__ bias, float* __restrict__ C, int M) {
  const int wave = threadIdx.x >> 5;
  const int lane = threadIdx.x & 31;
  constexpr int tilesN = N >> 4;
  const int tile = blockIdx.x * (blockDim.x >> 5) + wave;
  if (tile >= (M >> 4) * tilesN) return;   // wave-uniform: EXEC stays all-1s
  const int tm = (tile / tilesN) << 4;
  const int tn = (tile % tilesN) << 4;
  const int lmod = lane & 15;
  const int lhi  = lane >> 4;

  const float* aptr = A + (size_t)(tm + lmod) * K + 2 * lhi;
  const float* bptr = B + (size_t)(2 * lhi) * N + tn + lmod;
  v8f acc = {};
#pragma unroll 8
  for (int k = 0; k < K; k += 4) {
    v2f a = *(const v2f*)(aptr + k);                 // global_load_b64, imm off
    v2f b;
    b.x = bptr[(size_t)k * N];                       // imm offsets (k, N const)
    b.y = bptr[(size_t)(k + 1) * N];
    // (neg_a, A, neg_b, B, c_mod, C, reuse_a, reuse_b)
    acc = __builtin_amdgcn_wmma_f32_16x16x4_f32(
        false, a, false, b, (short)0, acc, false, false);
  }
  const float bv = bias[tn + lmod];
  float* cptr = C + (size_t)(tm + 8 * lhi) * N + tn + lmod;
#pragma unroll
  for (int r = 0; r < 8; ++r)
    cptr[(size_t)r * N] = acc[r] + bv;
}

// out[n] = (1/M) * [ sum_m o_m * sum_d tanh(ap[n,d]+bp[m,d])*w[d] + b_score*sum_m o_m ]
// One wave per 2 actors; lane covers d in {4l..4l+3} U {128+4l..128+4l+3}.
// Pairwise tree sums + split (lo/hi) accumulators: 4 independent carried FMA
// chains, single butterfly reduction at the end.
__global__ __launch_bounds__(256) void score_reduce(
    const float* __restrict__ ap,    // [N,256] (bias folded in)
    const float* __restrict__ bp,    // [M,256] (bias folded in)
    const float* __restrict__ outc,  // [M]
    const float* __restrict__ w,     // [256]
    const float* __restrict__ bsp,   // [1]
    float* __restrict__ out, int N, int M) {
  const int wave = threadIdx.x >> 5;
  const int lane = threadIdx.x & 31;
  const int n0 = (blockIdx.x * (blockDim.x >> 5) + wave) * 2;
  if (n0 >= N) return;
  const int d0 = 4 * lane;
  const int d1 = 128 + 4 * lane;

  const float4 w0 = *(const float4*)(w + d0);
  const float4 w1 = *(const float4*)(w + d1);
  const float4 aA0 = *(const float4*)(ap + (size_t)n0 * 256 + d0);
  const float4 aA1 = *(const float4*)(ap + (size_t)n0 * 256 + d1);
  const float4 aB0 = *(const float4*)(ap + (size_t)(n0 + 1) * 256 + d0);
  const float4 aB1 = *(const float4*)(ap + (size_t)(n0 + 1) * 256 + d1);

  float accA0 = 0.f, accA1 = 0.f, accB0 = 0.f, accB1 = 0.f, osum = 0.f;
#pragma unroll 2
  for (int m = 0; m < M; ++m) {
    const float4 b0 = *(const float4*)(bp + (size_t)m * 256 + d0);
    const float4 b1 = *(const float4*)(bp + (size_t)m * 256 + d1);
    const float o = outc[m];             // uniform broadcast load
    osum += o;

    // actor A, low half (tree sum: 2-deep adds, all tanh independent)
    float sA0 = (fast_tanhf(aA0.x + b0.x) * w0.x +
                 fast_tanhf(aA0.y + b0.y) * w0.y) +
                (fast_tanhf(aA0.z + b0.z) * w0.z +
                 fast_tanhf(aA0.w + b0.w) * w0.w);
    // actor A, high half
    float sA1 = (fast_tanhf(aA1.x + b1.x) * w1.x +
                 fast_tanhf(aA1.y + b1.y) * w1.y) +
                (fast_tanhf(aA1.z + b1.z) * w1.z +
                 fast_tanhf(aA1.w + b1.w) * w1.w);
    // actor B, low half
    float sB0 = (fast_tanhf(aB0.x + b0.x) * w0.x +
                 fast_tanhf(aB0.y + b0.y) * w0.y) +
                (fast_tanhf(aB0.z + b0.z) * w0.z +
                 fast_tanhf(aB0.w + b0.w) * w0.w);
    // actor B, high half
    float sB1 = (fast_tanhf(aB1.x + b1.x) * w1.x +
                 fast_tanhf(aB1.y + b1.y) * w1.y) +
                (fast_tanhf(aB1.z + b1.z) * w1.z +
                 fast_tanhf(aB1.w + b1.w) * w1.w);

    accA0 = fmaf(sA0, o, accA0);
    accA1 = fmaf(sA1, o, accA1);
    accB0 = fmaf(sB0, o, accB0);
    accB1 = fmaf(sB1, o, accB1);
  }

  float accA = accA0 + accA1;
  float accB = accB0 + accB1;
#pragma unroll
  for (int off = 16; off >= 1; off >>= 1) {
    accA += __shfl_xor(accA, off);
    accB += __shfl_xor(accB, off);
  }
  if (lane == 0) {
    const float bs  = bsp[0];
    const float inv = 1.0f / (float)M;
    out[n0]     = (accA + bs * osum) * inv;
    out[n0 + 1] = (accB + bs * osum) * inv;
  }
}

extern "C" void kernel_launch(void* const* d_in, const int* in_sizes, int n_in,
                              void* d_out, int out_size, void* d_ws, size_t ws_size,
                              hipStream_t stream) {
  const float* actor = (const float*)d_in[0];  // [2048,256]
  const float* bill  = (const float*)d_in[1];  // [1024,512]
  const float* outc  = (const float*)d_in[2];  // [1024]
  const float* W_a   = (const float*)d_in[3];  // [256,256]
  const float* b_a   = (const float*)d_in[4];  // [256]
  const float* W_b   = (const float*)d_in[5];  // [512,256]
  const float* b_b   = (const float*)d_in[6];  // [256]
  const float* w_s   = (const float*)d_in[7];  // [256]
  const float* b_s   = (const float*)d_in[8];  // [1]
  float* out = (float*)d_out;

  const int N = 2048, Mb = 1024, D = 256;
  float* ap = (float*)d_ws;             // N*D fp32  (2 MB)
  float* bp = ap + (size_t)N * D;       // Mb*D fp32 (1 MB)

  // actor_proj = actor @ W_actor + b_actor   (K=256)
  {
    const int tiles = (N / 16) * (D / 16);         // 2048 wave-tiles
    proj_gemm_wmma<256, 256><<<(tiles + 3) / 4, 128, 0, stream>>>(actor, W_a, b_a, ap, N);
  }
  // bill_proj = bill @ W_bill + b_bill       (K=512)
  {
    const int tiles = (Mb / 16) * (D / 16);        // 1024 wave-tiles
    proj_gemm_wmma<256, 512><<<(tiles + 3) / 4, 128, 0, stream>>>(bill, W_b, b_b, bp, Mb);
  }
  // scoring: 8 waves/block * 2 actors/wave = 16 actors per block
  score_reduce<<<N / 16, 256, 0, stream>>>(ap, bp, outc, w_s, b_s, out, N, Mb);
}